// NNTDLambda_84636625535230
// MI455X (gfx1250) — compile-verified
//
#include <hip/hip_runtime.h>
#include <hip/hip_bf16.h>

typedef float v2f __attribute__((ext_vector_type(2)));
typedef float v8f __attribute__((ext_vector_type(8)));

#define T_DIM 2048
#define D_DIM 256
#define H_DIM 256
#define GAMMA_LAMBDA (0.99f * 0.9f)
#define CONT_ALPHA 0.01f

// ---------------------------------------------------------------------------
// Kernel 1: H = relu(X @ W1 + b1)   (M=2048, N=256, K=256) via fp32 WMMA.
// One wave per 16x16 output tile; K loop in steps of 4 (V_WMMA_F32_16X16X4_F32).
// ---------------------------------------------------------------------------
__global__ __launch_bounds__(256) void fwd_gemm_kernel(
    const float* __restrict__ X, const float* __restrict__ W1,
    const float* __restrict__ b1, float* __restrict__ h)
{
    const int lane = threadIdx.x & 31;
    const int wave = threadIdx.x >> 5;
    const int tile = blockIdx.x * 8 + wave;   // 2048 tiles = 128(M) x 16(N)
    const int tileM = tile >> 4;
    const int tileN = tile & 15;
    const int half = lane >> 4;               // selects K pair {0,1} vs {2,3}
    const int l16  = lane & 15;
    const int rowA = tileM * 16 + l16;        // M index held by this lane (A)
    const int colB = tileN * 16 + l16;        // N index held by this lane (B/C/D)

    v8f acc = {};
    for (int k0 = 0; k0 < D_DIM; k0 += 4) {
        const int ka = k0 + 2 * half;
        v2f a, b;
        a.x = X[rowA * D_DIM + ka];
        a.y = X[rowA * D_DIM + ka + 1];
        b.x = W1[ka * H_DIM + colB];
        b.y = W1[(ka + 1) * H_DIM + colB];
        acc = __builtin_amdgcn_wmma_f32_16x16x4_f32(
            false, a, false, b, (short)0, acc, false, false);
    }
    const float bias = b1[colB];
#pragma unroll
    for (int v = 0; v < 8; ++v) {
        const int m = tileM * 16 + half * 8 + v;   // C/D layout: lanes>=16 hold M+8
        const float val = acc[v] + bias;
        h[m * H_DIM + colB] = val > 0.0f ? val : 0.0f;
    }
}

// ---------------------------------------------------------------------------
// Kernel 2: u[t] = dot(h[t,:], W2) + b2. One block per row, LDS tree reduce.
// ---------------------------------------------------------------------------
__global__ __launch_bounds__(256) void value_kernel(
    const float* __restrict__ h, const float* __restrict__ W2,
    const float* __restrict__ b2, float* __restrict__ u)
{
    __shared__ float s[256];
    const int t = blockIdx.x;
    const int j = threadIdx.x;
    s[j] = h[t * H_DIM + j] * W2[j];
    __syncthreads();
    for (int off = 128; off > 0; off >>= 1) {
        if (j < off) s[j] += s[j + off];
        __syncthreads();
    }
    if (j == 0) u[t] = s[0] + b2[0];
}

// ---------------------------------------------------------------------------
// Kernel 3: sequential scalar scans (forward for sigma/r, backward for w).
// Also emits total_b2 = -sum(w) and sigma_last. Trivial O(T) work: 1 thread.
// ---------------------------------------------------------------------------
__global__ void scan_kernel(const float* __restrict__ u, const float* __restrict__ ys,
                            float* __restrict__ sigma, float* __restrict__ w,
                            float* __restrict__ out)
{
    if (threadIdx.x != 0 || blockIdx.x != 0) return;
    float r = 0.0f;
    for (int t = 0; t < T_DIM; ++t) {
        const float g    = ys[2 * t + 0];
        const float done = ys[2 * t + 1];
        const float d    = g - u[t];
        const float sg   = d * d - r;
        sigma[t] = sg;
        r = r + CONT_ALPHA * sg;
        if (done > 0.5f) r = 0.0f;
    }
    float wnext = 0.0f;
    float sumw  = 0.0f;
    for (int t = T_DIM - 1; t >= 0; --t) {
        const float done = ys[2 * t + 1];
        const float keep = (done > 0.5f) ? 0.0f : 1.0f;
        const float wt = sigma[t] + GAMMA_LAMBDA * keep * wnext;
        w[t] = wt;
        wnext = wt;
        sumw += wt;
    }
    out[D_DIM * H_DIM + 2 * H_DIM + 0] = -sumw;            // total_b2
    out[D_DIM * H_DIM + 2 * H_DIM + 1] = sigma[T_DIM - 1]; // sigma_last
}

// ---------------------------------------------------------------------------
// Kernel 4: per column j:
//   Bs[t,j] = w[t] * ((h[t,j] > 0) ? W2[j] : 0)   (written in place over h)
//   total_b1[j] = -sum_t Bs[t,j]
//   total_W2[j] = -sum_t w[t] * h[t,j]
// ---------------------------------------------------------------------------
__global__ __launch_bounds__(256) void col_kernel(
    float* __restrict__ hBs, const float* __restrict__ W2,
    const float* __restrict__ w, float* __restrict__ out)
{
    __shared__ float s1[256];
    __shared__ float s2[256];
    const int j   = blockIdx.x;
    const int tid = threadIdx.x;
    const float w2j = W2[j];
    float a1 = 0.0f, a2 = 0.0f;
    for (int t = tid; t < T_DIM; t += 256) {
        const float hv = hBs[t * H_DIM + j];
        const float wt = w[t];
        const float bs = (hv > 0.0f) ? wt * w2j : 0.0f;
        hBs[t * H_DIM + j] = bs;      // in-place: hv consumed before overwrite
        a1 += bs;
        a2 += wt * hv;
    }
    s1[tid] = a1; s2[tid] = a2;
    __syncthreads();
    for (int off = 128; off > 0; off >>= 1) {
        if (tid < off) { s1[tid] += s1[tid + off]; s2[tid] += s2[tid + off]; }
        __syncthreads();
    }
    if (tid == 0) {
        out[D_DIM * H_DIM + j]         = -s1[0];  // total_b1
        out[D_DIM * H_DIM + H_DIM + j] = -s2[0];  // total_W2
    }
}

// ---------------------------------------------------------------------------
// Kernel 5: total_W1 = -(X^T @ Bs)   (M=256, N=256, K=2048) via fp32 WMMA.
// ---------------------------------------------------------------------------
__global__ __launch_bounds__(256) void bwd_gemm_kernel(
    const float* __restrict__ X, const float* __restrict__ Bs,
    float* __restrict__ out)
{
    const int lane = threadIdx.x & 31;
    const int wave = threadIdx.x >> 5;
    const int tile = blockIdx.x * 8 + wave;   // 256 tiles = 16(M) x 16(N)
    const int tileM = tile >> 4;
    const int tileN = tile & 15;
    const int half = lane >> 4;
    const int l16  = lane & 15;
    const int rowA = tileM * 16 + l16;        // i (D index)
    const int colB = tileN * 16 + l16;        // j (H index)

    v8f acc = {};
    for (int k0 = 0; k0 < T_DIM; k0 += 4) {
        const int ka = k0 + 2 * half;
        v2f a, b;
        a.x = X[ka * D_DIM + rowA];           // A[m,k] = X[k,m]  (X^T)
        a.y = X[(ka + 1) * D_DIM + rowA];
        b.x = Bs[ka * H_DIM + colB];
        b.y = Bs[(ka + 1) * H_DIM + colB];
        acc = __builtin_amdgcn_wmma_f32_16x16x4_f32(
            false, a, false, b, (short)0, acc, false, false);
    }
#pragma unroll
    for (int v = 0; v < 8; ++v) {
        const int m = tileM * 16 + half * 8 + v;
        out[m * H_DIM + colB] = -acc[v];      // total_W1 (row-major D x H)
    }
}

// ---------------------------------------------------------------------------
extern "C" void kernel_launch(void* const* d_in, const int* in_sizes, int n_in,
                              void* d_out, int out_size, void* d_ws, size_t ws_size,
                              hipStream_t stream)
{
    const float* xs = (const float*)d_in[0];   // (T, D)
    const float* ys = (const float*)d_in[1];   // (T, 2) = [g, done]
    const float* W1 = (const float*)d_in[2];   // (D, H)
    const float* b1 = (const float*)d_in[3];   // (H,)
    const float* W2 = (const float*)d_in[4];   // (H, 1)
    const float* b2 = (const float*)d_in[5];   // (1,)
    float* out = (float*)d_out;

    float* ws = (float*)d_ws;
    float* h     = ws;                       // T*H floats (reused as Bs in place)
    float* u     = h + (size_t)T_DIM * H_DIM;   // T
    float* sigma = u + T_DIM;                   // T
    float* w     = sigma + T_DIM;               // T

    fwd_gemm_kernel<<<256, 256, 0, stream>>>(xs, W1, b1, h);
    value_kernel<<<T_DIM, 256, 0, stream>>>(h, W2, b2, u);
    scan_kernel<<<1, 32, 0, stream>>>(u, ys, sigma, w, out);
    col_kernel<<<H_DIM, 256, 0, stream>>>(h, W2, w, out);
    bwd_gemm_kernel<<<32, 256, 0, stream>>>(xs, h, out);
}